// MultiShiftCausalMHSA_84361747628599
// MI455X (gfx1250) — compile-verified
//
#include <hip/hip_runtime.h>
#include <math.h>

// Problem constants (from reference)
#define Bb 4
#define Tt 1024
#define Dd 256
#define Hh 4
#define DHd 64          // head dim
#define NSH 4           // shifts {0,4,8,12}
#define WIN 64

typedef __attribute__((ext_vector_type(2))) float v2f;
typedef __attribute__((ext_vector_type(8))) float v8f;

__device__ __forceinline__ v8f wmma_f32(v2f a, v2f b, v8f c) {
  // D = A(16x4,f32) * B(4x16,f32) + C(16x16,f32)
  return __builtin_amdgcn_wmma_f32_16x16x4_f32(false, a, false, b, (short)0, c,
                                               false, false);
}

// ---------------------------------------------------------------------------
// C = A(4096x256) @ W^T(256x256) + bias.
// mode 0: scatter to (B,H,T,DH) layout (q/k/v). mode 1: row-major (B,T,D).
// Block: 256 threads (8 waves). Grid: 256 m-tiles. Wave computes two 16x16 tiles.
// ---------------------------------------------------------------------------
__global__ void gemm_qkvo(const float* __restrict__ A, const float* __restrict__ W,
                          const float* __restrict__ bias, float* __restrict__ out,
                          int mode) {
  __shared__ float xs[16 * Dd];   // 16 KB: one 16x256 A tile
  const int m0 = blockIdx.x * 16;

  for (int i = threadIdx.x; i < 16 * Dd; i += blockDim.x)
    xs[i] = A[(size_t)m0 * Dd + i];
  __syncthreads();

  const int wave = threadIdx.x >> 5;
  const int lane = threadIdx.x & 31;
  const int half = lane >> 4;
  const int col  = lane & 15;

  for (int nt = wave; nt < 16; nt += 8) {
    const int n0 = nt * 16;
    const int n  = n0 + col;            // this lane's B column / output feature
    v8f c = {};
#pragma unroll 4
    for (int kk = 0; kk < Dd; kk += 4) {
      const int ka = kk + 2 * half;     // even -> 8B aligned
      v2f a = *(const v2f*)&xs[col * Dd + ka];         // A[col, ka..ka+1]
      v2f b = *(const v2f*)&W[(size_t)n * Dd + ka];    // B[ka..ka+1, n] = W[n, ka..]
      c = wmma_f32(a, b, c);
    }
    const float bv = bias[n];
    if (mode == 0) {
      const int h = n >> 6, d = n & 63;
#pragma unroll
      for (int r = 0; r < 8; ++r) {
        const int m = m0 + r + 8 * half;        // token index
        const int b_ = m >> 10, t = m & (Tt - 1);
        out[(((size_t)b_ * Hh + h) * Tt + t) * DHd + d] = c[r] + bv;
      }
    } else {
#pragma unroll
      for (int r = 0; r < 8; ++r) {
        const int m = m0 + r + 8 * half;
        out[(size_t)m * Dd + n] = c[r] + bv;
      }
    }
  }
}

// ---------------------------------------------------------------------------
// Banded multi-shift attention. Grid: (T/16, B*H). Block: 32 (one wave).
// For query tile t0..t0+15: band of keys j in [t0-16, t0+80) (6 subtiles of 16).
// ---------------------------------------------------------------------------
__global__ void attn_kernel(const float* __restrict__ q, const float* __restrict__ k0,
                            const float* __restrict__ v0,
                            const float* __restrict__ shift_logits,
                            const float* __restrict__ alibi_slope,
                            float* __restrict__ ctx) {
  __shared__ float s0[16 * 96];   // raw scores vs unshifted k0
  __shared__ float pS[16 * 96];   // softmax probabilities, col = k - t0
  __shared__ float vs[96 * DHd];  // Vsum band (sum over shifts)

  const int t0 = blockIdx.x * 16;
  const int bh = blockIdx.y;
  const int b_ = bh >> 2, h = bh & 3;
  const float* qh = q  + (size_t)bh * Tt * DHd;
  const float* kh = k0 + (size_t)bh * Tt * DHd;
  const float* vh = v0 + (size_t)bh * Tt * DHd;

  const int lane = threadIdx.x;
  const int half = lane >> 4;
  const int col  = lane & 15;

  // log(softmax(shift_logits[h]) + 1e-8)
  float sl[NSH], lp[NSH];
#pragma unroll
  for (int s = 0; s < NSH; ++s) sl[s] = shift_logits[h * NSH + s];
  {
    float m = fmaxf(fmaxf(sl[0], sl[1]), fmaxf(sl[2], sl[3]));
    float e[NSH], sum = 0.f;
#pragma unroll
    for (int s = 0; s < NSH; ++s) { e[s] = __expf(sl[s] - m); sum += e[s]; }
#pragma unroll
    for (int s = 0; s < NSH; ++s) lp[s] = __logf(e[s] / sum + 1e-8f);
  }
  const float slope = alibi_slope[h];

  // Preload scaled Q fragments: 16 k-steps of (16x4)
  v2f qa[16];
#pragma unroll
  for (int kk = 0; kk < 16; ++kk) {
    const int d = kk * 4 + 2 * half;
    v2f t = *(const v2f*)&qh[(size_t)(t0 + col) * DHd + d];
    qa[kk].x = t.x * 0.125f;   // 1/sqrt(64)
    qa[kk].y = t.y * 0.125f;
  }

  // S0[r, c] = qn[t0+r] . k0[t0-16+c]   (zero rows for j<0 / j>=T)
  for (int jt = 0; jt < 6; ++jt) {
    const int j = t0 - 16 + jt * 16 + col;        // this lane's key row
    const float mk = (j >= 0 && j < Tt) ? 1.f : 0.f;
    const int jc = j < 0 ? 0 : (j >= Tt ? Tt - 1 : j);
    v8f c = {};
#pragma unroll 4
    for (int kk = 0; kk < 16; ++kk) {
      const int d = kk * 4 + 2 * half;
      v2f bf = *(const v2f*)&kh[(size_t)jc * DHd + d];
      bf.x *= mk; bf.y *= mk;
      c = wmma_f32(qa[kk], bf, c);
    }
#pragma unroll
    for (int r = 0; r < 8; ++r)
      s0[(r + 8 * half) * 96 + jt * 16 + col] = c[r];
  }
  __syncthreads();

  // Combine shifts (logsumexp) + ALiBi + softmax over the 64-wide band.
  // Lane pair (2r, 2r+1) owns query row r; each lane does 32 of 64 key offsets.
  const int r    = lane >> 1;
  const int wpar = lane & 1;
  float attv[32];
  float mx = -3.4e38f;
#pragma unroll
  for (int i = 0; i < 32; ++i) {
    const int w = wpar + 2 * i;            // rel = k - t in [0,64)
    const int kpos = t0 + r + w;
    float a;
    if (kpos < Tt) {
      float t0v = s0[r * 96 + (r + w + 16 - 0)]  + lp[0];
      float t1v = s0[r * 96 + (r + w + 16 - 4)]  + lp[1];
      float t2v = s0[r * 96 + (r + w + 16 - 8)]  + lp[2];
      float t3v = s0[r * 96 + (r + w + 16 - 12)] + lp[3];
      float m2 = fmaxf(fmaxf(t0v, t1v), fmaxf(t2v, t3v));
      a = m2 + __logf(__expf(t0v - m2) + __expf(t1v - m2) +
                      __expf(t2v - m2) + __expf(t3v - m2));
      a -= slope * (float)w;               // alibi; window mask = 0 in band
    } else {
      a = -3.0e38f;                        // key beyond sequence
    }
    attv[i] = a;
    mx = fmaxf(mx, a);
  }
  mx = fmaxf(mx, __shfl_xor(mx, 1, 32));
  float sum = 0.f;
#pragma unroll
  for (int i = 0; i < 32; ++i) { attv[i] = __expf(attv[i] - mx); sum += attv[i]; }
  sum += __shfl_xor(sum, 1, 32);
  const float inv = 1.0f / sum;

  for (int i = lane; i < 16 * 96; i += 32) pS[i] = 0.f;
  __syncthreads();
#pragma unroll
  for (int i = 0; i < 32; ++i) {
    const int w = wpar + 2 * i;
    pS[r * 96 + (r + w)] = attv[i] * inv;  // col = k - t0 in [0,79)
  }

  // Vsum band: vs[c, d] = sum_s v0[t0 + c - 4s, d]
  for (int idx = lane; idx < 96 * DHd; idx += 32) {
    const int kc = idx >> 6, d = idx & 63;
    float acc = 0.f;
#pragma unroll
    for (int s = 0; s < NSH; ++s) {
      const int jj = t0 + kc - 4 * s;
      const float mk = (jj >= 0 && jj < Tt) ? 1.f : 0.f;
      const int jc = jj < 0 ? 0 : (jj >= Tt ? Tt - 1 : jj);
      acc += vh[(size_t)jc * DHd + d] * mk;
    }
    vs[idx] = acc;
  }
  __syncthreads();

  // ctx(16x64) = pS(16x96) @ vs(96x64), written as (B,T,D) rows for final GEMM
#pragma unroll
  for (int nt = 0; nt < 4; ++nt) {
    const int n = nt * 16 + col;
    v8f c = {};
#pragma unroll 4
    for (int kk = 0; kk < 96; kk += 4) {
      const int kr = kk + 2 * half;
      v2f a = *(const v2f*)&pS[r * 0 + col * 96 + kr];   // P[col, kr..kr+1]
      v2f bf;
      bf.x = vs[(kr + 0) * DHd + n];
      bf.y = vs[(kr + 1) * DHd + n];
      c = wmma_f32(a, bf, c);
    }
#pragma unroll
    for (int rr = 0; rr < 8; ++rr) {
      const int t = t0 + rr + 8 * half;
      ctx[((size_t)b_ * Tt + t) * Dd + h * DHd + n] = c[rr];
    }
  }
}

// ---------------------------------------------------------------------------
extern "C" void kernel_launch(void* const* d_in, const int* in_sizes, int n_in,
                              void* d_out, int out_size, void* d_ws, size_t ws_size,
                              hipStream_t stream) {
  (void)in_sizes; (void)n_in; (void)out_size; (void)ws_size;
  const float* x   = (const float*)d_in[0];
  const float* Wq  = (const float*)d_in[1];
  const float* bq  = (const float*)d_in[2];
  const float* Wk  = (const float*)d_in[3];
  const float* bk  = (const float*)d_in[4];
  const float* Wv  = (const float*)d_in[5];
  const float* bv  = (const float*)d_in[6];
  const float* Wo  = (const float*)d_in[7];
  const float* bo  = (const float*)d_in[8];
  const float* shift_logits = (const float*)d_in[9];
  const float* alibi_slope  = (const float*)d_in[10];
  float* out = (float*)d_out;

  float* ws  = (float*)d_ws;
  const size_t perBuf = (size_t)Bb * Hh * Tt * DHd;  // 1,048,576 floats = 4 MB
  float* q   = ws;
  float* k   = ws + perBuf;
  float* v   = ws + 2 * perBuf;
  float* ctx = ws + 3 * perBuf;

  // QKV projections (WMMA fp32), scattered to (B,H,T,DH)
  gemm_qkvo<<<dim3(Bb * Tt / 16), 256, 0, stream>>>(x, Wq, bq, q, 0);
  gemm_qkvo<<<dim3(Bb * Tt / 16), 256, 0, stream>>>(x, Wk, bk, k, 0);
  gemm_qkvo<<<dim3(Bb * Tt / 16), 256, 0, stream>>>(x, Wv, bv, v, 0);

  // Banded multi-shift attention -> ctx (B,T,D)
  attn_kernel<<<dim3(Tt / 16, Bb * Hh), 32, 0, stream>>>(q, k, v, shift_logits,
                                                         alibi_slope, ctx);

  // Output projection
  gemm_qkvo<<<dim3(Bb * Tt / 16), 256, 0, stream>>>(ctx, Wo, bo, out, 1);
}